// Actor_82471962018780
// MI455X (gfx1250) — compile-verified
//
#include <hip/hip_runtime.h>

#define NN    1024
#define SS    1025      // N + 1 (depot prepended)
#define HH    256
#define AA    256
#define EE    16
#define GH    1024      // 4*H
#define APAD  260       // A-tile LDS row stride (floats): 256+4 kills bank conflicts

typedef float v2f __attribute__((ext_vector_type(2)));
typedef float v8f __attribute__((ext_vector_type(8)));

#if defined(__has_builtin)
#if __has_builtin(__builtin_amdgcn_global_load_async_to_lds_b128)
#define HAVE_ASYNC_LDS 1
#endif
#endif

#ifdef HAVE_ASYNC_LDS
// Param type per hipcc diagnostic: 'int __attribute__((vector_size(16))) __device__ *'
typedef int v4i_g __attribute__((vector_size(16)));
typedef __attribute__((address_space(1))) v4i_g gbl_v4i;   // prints as __device__
typedef __attribute__((address_space(3))) v4i_g lds_v4i;   // prints as __shared__
#endif

__device__ __forceinline__ void wait_async_all() {
#if defined(__has_builtin) && __has_builtin(__builtin_amdgcn_s_wait_asynccnt)
    __builtin_amdgcn_s_wait_asynccnt(0);
#else
    asm volatile("s_wait_asynccnt 0x0" ::: "memory");
#endif
}

__device__ __forceinline__ float fast_sigmoid(float x) {
    return 1.0f / (1.0f + __expf(-x));
}
__device__ __forceinline__ float fast_tanh(float x) {
    float e = __expf(2.0f * x);
    return 1.0f - 2.0f / (e + 1.0f);
}

// ---------------------------------------------------------------------------
// Encoder LSTM scan: one persistent 1024-thread workgroup (32 waves on a WGP).
// Thread tid owns gate row tid (of 4H=1024). h/c/gates live in LDS; weights
// stream from L2 (enc_Whh = 1MB, L2-resident).
// ---------------------------------------------------------------------------
__global__ void __launch_bounds__(1024) encoder_lstm(
    const float* __restrict__ items,   // [1,N,2]
    const float* __restrict__ W_emb,   // [E,2]
    const float* __restrict__ Wih,     // [4H,E]
    const float* __restrict__ Whh,     // [4H,H]
    const float* __restrict__ bih,     // [4H]
    const float* __restrict__ bhh,     // [4H]
    float* __restrict__ enc_out,       // [S,H]
    float* __restrict__ hc_final)      // [2H]
{
    __shared__ __align__(16) float xs[EE];
    __shared__ __align__(16) float hs[HH];
    __shared__ __align__(16) float cs[HH];
    __shared__ __align__(16) float gs[GH];
    const int tid = threadIdx.x;
    if (tid < HH) { hs[tid] = 0.0f; cs[tid] = 0.0f; }
    const float bsum = bih[tid] + bhh[tid];
    const float4* __restrict__ wih4 = reinterpret_cast<const float4*>(Wih + tid * EE);
    const float4* __restrict__ whh4 = reinterpret_cast<const float4*>(Whh + tid * HH);
    __builtin_prefetch(Whh + tid * HH, 0, 1);   // global_prefetch_b8
    __syncthreads();

    for (int t = 0; t < SS; ++t) {
        if (tid < EE) {
            float ix = 0.0f, iy = 0.0f;
            if (t > 0) { ix = items[(t - 1) * 2 + 0]; iy = items[(t - 1) * 2 + 1]; }
            xs[tid] = W_emb[tid * 2 + 0] * ix + W_emb[tid * 2 + 1] * iy;
        }
        __syncthreads();
        float acc = bsum;
        const float4* x4 = reinterpret_cast<const float4*>(xs);
        #pragma unroll
        for (int k = 0; k < EE / 4; ++k) {
            float4 w = wih4[k]; float4 v = x4[k];
            acc += w.x * v.x + w.y * v.y + w.z * v.z + w.w * v.w;
        }
        const float4* h4 = reinterpret_cast<const float4*>(hs);
        #pragma unroll 4
        for (int k = 0; k < HH / 4; ++k) {
            float4 w = whh4[k]; float4 v = h4[k];
            acc += w.x * v.x + w.y * v.y + w.z * v.z + w.w * v.w;
        }
        gs[tid] = acc;
        __syncthreads();
        if (tid < HH) {
            float ig = fast_sigmoid(gs[tid]);
            float fg = fast_sigmoid(gs[HH + tid]);
            float gg = fast_tanh(gs[2 * HH + tid]);
            float og = fast_sigmoid(gs[3 * HH + tid]);
            float c2 = fg * cs[tid] + ig * gg;
            float h2 = og * fast_tanh(c2);
            cs[tid] = c2; hs[tid] = h2;
            enc_out[t * HH + tid] = h2;
        }
        __syncthreads();
    }
    if (tid < HH) { hc_final[tid] = hs[tid]; hc_final[HH + tid] = cs[tid]; }
}

// ---------------------------------------------------------------------------
// enc_proj = enc_out @ W1.T  ([1025,256] x [256,256]) with V_WMMA_F32_16X16X4_F32.
// Block = 256 threads (8 waves) owning one 16-row tile x 8 column tiles.
// The shared A tile (16x256 f32) is staged into LDS once per block via the
// gfx1250 async global->LDS path (ASYNCcnt + s_wait_asynccnt), row stride
// padded to 260 floats so the 16-lane WMMA A-reads hit 16 distinct banks.
// B streams from L2. Row guard is a scalar branch (only mt==64 is ragged).
// ---------------------------------------------------------------------------
__global__ void __launch_bounds__(256) enc_proj_wmma(
    const float* __restrict__ enc_out,  // [S,H]
    const float* __restrict__ W1,       // [A,H]
    float* __restrict__ enc_proj)       // [S,A]
{
    __shared__ __align__(16) float Atile[16 * APAD];   // ~16.6 KB
    const int tid  = threadIdx.x;
    const int wv   = tid >> 5;               // 0..7
    const int lane = tid & 31;
    const int mt   = blockIdx.x >> 1;        // 0..64 row tile
    const int half = blockIdx.x & 1;
    const int nt   = half * 8 + wv;          // 0..15 column tile
    const int m16  = lane & 15;
    const int hi   = lane >> 4;

    // --- stage A tile: 16 rows x 64 float4 segments, 4 per thread
    #pragma unroll
    for (int i = 0; i < 4; ++i) {
        const int cidx = i * 256 + tid;
        const int r    = cidx >> 6;          // 0..15
        const int seg  = cidx & 63;          // float4 within row
        const int grow = mt * 16 + r;
        const int growc = (grow < SS) ? grow : 0;      // clamp (dup, never OOB)
        const float* src = enc_out + growc * HH + seg * 4;
        float*       dst = &Atile[r * APAD + seg * 4];
#ifdef HAVE_ASYNC_LDS
        __builtin_amdgcn_global_load_async_to_lds_b128(
            (gbl_v4i*)src, (lds_v4i*)dst, 0, 0);
#else
        *reinterpret_cast<float4*>(dst) = *reinterpret_cast<const float4*>(src);
#endif
    }
#ifdef HAVE_ASYNC_LDS
    wait_async_all();
#endif
    __syncthreads();

    // --- WMMA K-loop: A from LDS, B (= W1 rows) from L2
    const int col = nt * 16 + m16;
    const float* __restrict__ arow = &Atile[m16 * APAD + 2 * hi];
    const float* __restrict__ brow = W1 + col * HH + 2 * hi;
    v8f acc = {};
    #pragma unroll 8
    for (int k = 0; k < HH; k += 4) {
        v2f a = *reinterpret_cast<const v2f*>(arow + k);   // A[m][k..k+1]
        v2f b = *reinterpret_cast<const v2f*>(brow + k);   // B[k..k+1][n] = W1[n][k..k+1]
        acc = __builtin_amdgcn_wmma_f32_16x16x4_f32(
                  /*neg_a=*/false, a, /*neg_b=*/false, b,
                  /*c_mod=*/(short)0, acc, /*reuse_a=*/false, /*reuse_b=*/false);
    }

    // --- epilogue: scalar-uniform guard, only the ragged last row tile branches
    if (mt < 64) {
        #pragma unroll
        for (int r = 0; r < 8; ++r)
            enc_proj[(mt * 16 + r + 8 * hi) * AA + col] = acc[r];
    } else {
        #pragma unroll
        for (int r = 0; r < 8; ++r) {
            const int orow = 1024 + r + 8 * hi;            // only r==0,hi==0 valid
            if (orow < SS) enc_proj[orow * AA + col] = acc[r];
        }
    }
}

// ---------------------------------------------------------------------------
// Decoder scan: one persistent 1024-thread workgroup. Per step:
//   1) w2h = W2@h          2) scores[s] = V . tanh(enc_proj[s]+w2h)
//   3) softmax -> aj       4) di = aj @ enc_out (4-way partials in LDS)
//   5) LSTM([di,dec2])     6) masked log-softmax -> probs row
//   7) masked argmax (first-index ties) -> pred, mask/dec2 update
// ---------------------------------------------------------------------------
__global__ void __launch_bounds__(1024) decoder_loop(
    const float* __restrict__ items,
    const float* __restrict__ dWih,    // [4H, H+2]
    const float* __restrict__ dWhh,    // [4H, H]
    const float* __restrict__ dbih,
    const float* __restrict__ dbhh,
    const float* __restrict__ W2,      // [A,H]
    const float* __restrict__ V,       // [A]
    const float* __restrict__ enc_out, // [S,H]
    const float* __restrict__ enc_proj,// [S,A]
    const float* __restrict__ hc_init, // [2H]
    float* __restrict__ probs,         // [S,S]
    int* __restrict__ preds)           // [S]
{
    __shared__ __align__(16) float hs[HH], cs[HH], w2h[AA], di[HH];
    __shared__ __align__(16) float scores[SS], ajv[SS], maskv[SS];
    __shared__ __align__(16) float gs[GH];     // gates; also di partials
    __shared__ __align__(16) float red[1024];
    __shared__ int redi[1024];
    __shared__ float dec2[2];
    const int tid = threadIdx.x;

    if (tid < HH) { hs[tid] = hc_init[tid]; cs[tid] = hc_init[HH + tid]; }
    for (int s = tid; s < SS; s += 1024) maskv[s] = (s == 0) ? 0.0f : 1.0f;
    if (tid == 0) { dec2[0] = 0.0f; dec2[1] = 0.0f; }
    const float bsum = dbih[tid] + dbhh[tid];
    const float2* __restrict__ wih2 = reinterpret_cast<const float2*>(dWih + tid * (HH + 2));
    const float4* __restrict__ whh4 = reinterpret_cast<const float4*>(dWhh + tid * HH);
    __syncthreads();

    for (int t = 0; t < SS; ++t) {
        // --- 1: w2h = W2 @ h (pre-update h)
        if (tid < AA) {
            const float4* w4 = reinterpret_cast<const float4*>(W2 + tid * HH);
            const float4* h4 = reinterpret_cast<const float4*>(hs);
            float acc = 0.0f;
            #pragma unroll 4
            for (int k = 0; k < HH / 4; ++k) {
                float4 w = w4[k]; float4 v = h4[k];
                acc += w.x * v.x + w.y * v.y + w.z * v.z + w.w * v.w;
            }
            w2h[tid] = acc;
        }
        __syncthreads();
        // --- 2: attention logits
        for (int s = tid; s < SS; s += 1024) {
            const float4* p4 = reinterpret_cast<const float4*>(enc_proj + s * AA);
            const float4* u4 = reinterpret_cast<const float4*>(w2h);
            const float4* v4 = reinterpret_cast<const float4*>(V);
            float acc = 0.0f;
            for (int k = 0; k < AA / 4; ++k) {
                float4 p = p4[k]; float4 u = u4[k]; float4 vv = v4[k];
                acc += vv.x * fast_tanh(p.x + u.x) + vv.y * fast_tanh(p.y + u.y)
                     + vv.z * fast_tanh(p.z + u.z) + vv.w * fast_tanh(p.w + u.w);
            }
            scores[s] = acc;
        }
        __syncthreads();
        // --- 3: softmax(scores) -> ajv
        float lmax = -3.0e38f;
        for (int s = tid; s < SS; s += 1024) lmax = fmaxf(lmax, scores[s]);
        red[tid] = lmax; __syncthreads();
        for (int off = 512; off > 0; off >>= 1) {
            if (tid < off) red[tid] = fmaxf(red[tid], red[tid + off]);
            __syncthreads();
        }
        const float smax = red[0]; __syncthreads();
        float lsum = 0.0f;
        for (int s = tid; s < SS; s += 1024) { float e = __expf(scores[s] - smax); ajv[s] = e; lsum += e; }
        red[tid] = lsum; __syncthreads();
        for (int off = 512; off > 0; off >>= 1) {
            if (tid < off) red[tid] += red[tid + off];
            __syncthreads();
        }
        const float inv = 1.0f / red[0]; __syncthreads();
        for (int s = tid; s < SS; s += 1024) ajv[s] *= inv;
        __syncthreads();
        // --- 4: di = aj @ enc_out (4 partial sums per column in gs)
        {
            const int part = tid >> 8;
            const int hh   = tid & 255;
            const int sBeg = part * 256;
            const int sEnd = (part == 3) ? SS : (sBeg + 256);
            float acc = 0.0f;
            for (int s = sBeg; s < sEnd; ++s) acc += ajv[s] * enc_out[s * HH + hh];
            gs[part * 256 + hh] = acc;
        }
        __syncthreads();
        if (tid < HH) di[tid] = gs[tid] + gs[256 + tid] + gs[512 + tid] + gs[768 + tid];
        __syncthreads();
        // --- 5: LSTM gates with x_in = [di, dec2]
        {
            float acc = bsum;
            const float2* d2 = reinterpret_cast<const float2*>(di);
            #pragma unroll 4
            for (int k = 0; k < HH / 2; ++k) {
                float2 w = wih2[k]; float2 v = d2[k];
                acc += w.x * v.x + w.y * v.y;
            }
            { float2 w = wih2[HH / 2]; acc += w.x * dec2[0] + w.y * dec2[1]; }
            const float4* h4 = reinterpret_cast<const float4*>(hs);
            #pragma unroll 4
            for (int k = 0; k < HH / 4; ++k) {
                float4 w = whh4[k]; float4 v = h4[k];
                acc += w.x * v.x + w.y * v.y + w.z * v.z + w.w * v.w;
            }
            gs[tid] = acc;
        }
        __syncthreads();
        if (tid < HH) {
            float ig = fast_sigmoid(gs[tid]);
            float fg = fast_sigmoid(gs[HH + tid]);
            float gg = fast_tanh(gs[2 * HH + tid]);
            float og = fast_sigmoid(gs[3 * HH + tid]);
            float c2 = fg * cs[tid] + ig * gg;
            cs[tid] = c2; hs[tid] = og * fast_tanh(c2);
        }
        __syncthreads();
        // --- 6: masked log-softmax -> probs row t (y stored in ajv)
        float ymax = -3.0e38f;
        for (int s = tid; s < SS; s += 1024) {
            float xm = scores[s] * maskv[s];
            float y  = (xm == 0.0f) ? -3.0e-37f : xm;
            ajv[s] = y;
            ymax = fmaxf(ymax, y);
        }
        red[tid] = ymax; __syncthreads();
        for (int off = 512; off > 0; off >>= 1) {
            if (tid < off) red[tid] = fmaxf(red[tid], red[tid + off]);
            __syncthreads();
        }
        const float m2 = red[0]; __syncthreads();
        float esum = 0.0f;
        for (int s = tid; s < SS; s += 1024) esum += __expf(ajv[s] - m2);
        red[tid] = esum; __syncthreads();
        for (int off = 512; off > 0; off >>= 1) {
            if (tid < off) red[tid] += red[tid + off];
            __syncthreads();
        }
        const float lse = __logf(red[0]); __syncthreads();
        for (int s = tid; s < SS; s += 1024) probs[(long)t * SS + s] = ajv[s] - m2 - lse;
        // --- 7: argmax of masked scores (first index on ties)
        float bv = -3.0e38f; int bi = 0;
        for (int s = tid; s < SS; s += 1024) {
            float xm = scores[s] * maskv[s];
            float z  = (xm == 0.0f) ? -3.0e37f : xm;
            if (z > bv) { bv = z; bi = s; }
        }
        red[tid] = bv; redi[tid] = bi; __syncthreads();
        for (int off = 512; off > 0; off >>= 1) {
            if (tid < off) {
                float v2 = red[tid + off]; int i2 = redi[tid + off];
                if (v2 > red[tid] || (v2 == red[tid] && i2 < redi[tid])) {
                    red[tid] = v2; redi[tid] = i2;
                }
            }
            __syncthreads();
        }
        const int pred = redi[0];
        __syncthreads();
        if (tid == 0) {
            maskv[pred] = 0.0f; maskv[0] = 1.0f;
            float dx = 0.0f, dy = 0.0f;
            if (pred > 0) { dx = items[(pred - 1) * 2 + 0]; dy = items[(pred - 1) * 2 + 1]; }
            dec2[0] = dx; dec2[1] = dy;
            preds[t] = pred;
        }
        __syncthreads();
    }
}

extern "C" void kernel_launch(void* const* d_in, const int* in_sizes, int n_in,
                              void* d_out, int out_size, void* d_ws, size_t ws_size,
                              hipStream_t stream)
{
    (void)in_sizes; (void)n_in; (void)out_size; (void)ws_size;
    const float* items = (const float*)d_in[0];
    const float* W_emb = (const float*)d_in[1];
    const float* eWih  = (const float*)d_in[2];
    const float* eWhh  = (const float*)d_in[3];
    const float* ebih  = (const float*)d_in[4];
    const float* ebhh  = (const float*)d_in[5];
    const float* dWih  = (const float*)d_in[6];
    const float* dWhh  = (const float*)d_in[7];
    const float* dbih  = (const float*)d_in[8];
    const float* dbhh  = (const float*)d_in[9];
    const float* W1    = (const float*)d_in[10];
    const float* W2    = (const float*)d_in[11];
    const float* V     = (const float*)d_in[12];
    // d_in[13] = budget (unused by the reference computation)

    float* ws       = (float*)d_ws;
    float* enc_out  = ws;                       // S*H floats
    float* enc_proj = enc_out + SS * HH;        // S*A floats
    float* hc       = enc_proj + SS * AA;       // 2H floats

    float* probs = (float*)d_out;                           // [S,S] f32 log-probs
    int*   preds = (int*)((float*)d_out + (size_t)SS * SS); // [S] int32

    encoder_lstm<<<1, 1024, 0, stream>>>(items, W_emb, eWih, eWhh, ebih, ebhh,
                                         enc_out, hc);
    enc_proj_wmma<<<65 * 2, 256, 0, stream>>>(enc_out, W1, enc_proj);
    decoder_loop<<<1, 1024, 0, stream>>>(items, dWih, dWhh, dbih, dbhh, W2, V,
                                         enc_out, enc_proj, hc, probs, preds);
}